// ROIFeatureExtraction_24086176596175
// MI455X (gfx1250) — compile-verified
//
#include <hip/hip_runtime.h>
#include <cstdint>

typedef __attribute__((ext_vector_type(2))) float v2f;
typedef __attribute__((ext_vector_type(8))) float v8f;

namespace {
constexpr int   kB    = 2;
constexpr int   kN    = 512;
constexpr int   kC    = 256;
constexpr int   kOut  = 14;          // OUT
constexpr int   kS    = 28;          // OUT*RATIO
constexpr int   kPix  = kOut * kOut; // 196
constexpr float kThr  = 0.5f;
constexpr int   kChk     = 8;        // channels per LDS staging round (== waves/block)
constexpr int   kMaxTWp  = 48;       // padded tile width cap for WMMA path
constexpr int   kMaxTHp  = 132;      // padded tile height cap for WMMA path
constexpr int   kSlot    = 2112;     // per-channel LDS tile slot (floats)
constexpr int   kWyPitch = 132;      // Wy row pitch (floats)
}

// ---------------------------------------------------------------- kernel 1
__global__ void roi_prep_kernel(const float* __restrict__ boxes,
                                float* __restrict__ verts,
                                int* __restrict__ level) {
    int i = blockIdx.x * blockDim.x + threadIdx.x;
    if (i >= kB * kN) return;
    const float* bx = boxes + (size_t)i * 4;
    float cx = bx[0], cy = bx[1], w = bx[2], h = bx[3];
    float lf = floorf(3.0f + log2f(sqrtf(w * h) * (1.0f / 224.0f)));
    lf = fminf(fmaxf(lf, 1.0f), 4.0f);
    level[i] = (int)lf;
    float* v = verts + (size_t)i * 4;
    v[0] = cx - 0.5f * w;
    v[1] = cy - 0.5f * h;
    v[2] = cx + 0.5f * w;
    v[3] = cy + 0.5f * h;
}

// ---------------------------------------------------------------- kernel 2
__global__ __launch_bounds__(512) void nms_kernel(
        const float* __restrict__ scores, const float* __restrict__ verts,
        const int* __restrict__ level, int* __restrict__ keep,
        float* __restrict__ out_boxes, float* __restrict__ out_keep) {
    const int b = blockIdx.x;
    const int t = threadIdx.x;
    __shared__ float s_sc[kN];
    __shared__ int   s_id[kN];
    __shared__ float s_x1[kN], s_y1[kN], s_x2[kN], s_y2[kN], s_ar[kN];
    __shared__ int   s_lv[kN];
    __shared__ unsigned char s_kp[kN];

    s_sc[t] = scores[b * kN + t];
    s_id[t] = t;
    __syncthreads();

    for (int k = 2; k <= kN; k <<= 1) {
        for (int j = k >> 1; j > 0; j >>= 1) {
            int ixj = t ^ j;
            if (ixj > t) {
                bool dir = ((t & k) == 0);
                float a = s_sc[t], c = s_sc[ixj];
                bool sw = dir ? (a < c) : (a > c);
                if (sw) {
                    s_sc[t] = c; s_sc[ixj] = a;
                    int ia = s_id[t]; s_id[t] = s_id[ixj]; s_id[ixj] = ia;
                }
            }
            __syncthreads();
        }
    }

    const int orig = s_id[t];
    const float* v = verts + ((size_t)b * kN + orig) * 4;
    float x1 = v[0], y1 = v[1], x2 = v[2], y2 = v[3];
    s_x1[t] = x1; s_y1[t] = y1; s_x2[t] = x2; s_y2[t] = y2;
    s_ar[t] = (x2 - x1) * (y2 - y1);
    s_lv[t] = level[b * kN + orig];
    s_kp[t] = 1;
    __syncthreads();

    for (int i = 0; i < kN - 1; ++i) {
        if (s_kp[i]) {
            if (t > i && s_kp[t] && s_lv[t] == s_lv[i]) {
                float ix1 = fmaxf(s_x1[i], s_x1[t]);
                float iy1 = fmaxf(s_y1[i], s_y1[t]);
                float ix2 = fminf(s_x2[i], s_x2[t]);
                float iy2 = fminf(s_y2[i], s_y2[t]);
                float inter = fmaxf(ix2 - ix1, 0.0f) * fmaxf(iy2 - iy1, 0.0f);
                float iou = inter / (s_ar[i] + s_ar[t] - inter + 1e-9f);
                if (iou > kThr) s_kp[t] = 0;
            }
        }
        __syncthreads();
    }

    int kp = s_kp[t];
    keep[b * kN + orig]     = kp;
    out_keep[b * kN + orig] = kp ? 1.0f : 0.0f;
    float* ob = out_boxes + ((size_t)b * kN + orig) * 4;
    ob[0] = kp ? x1 : 0.0f;
    ob[1] = kp ? y1 : 0.0f;
    ob[2] = kp ? x2 : 0.0f;
    ob[3] = kp ? y2 : 0.0f;
}

// ---------------------------------------------------------------- kernel 3
// Separable ROI-Align: pooled = Wy(14xTH) * tile(THxTW) * Wx^T(TWx14) per
// channel, executed with V_WMMA_F32_16X16X4_F32. Feature tiles are staged
// into LDS with gfx1250 async global->LDS loads (ASYNCcnt). One wave per
// channel; temp = Wy*tile round-trips through a per-wave LDS strip.
__global__ __launch_bounds__(256) void roi_align_kernel(
        const float* __restrict__ p4,  const float* __restrict__ p8,
        const float* __restrict__ p16, const float* __restrict__ p32,
        const float* __restrict__ verts, const int* __restrict__ level,
        const int* __restrict__ keep, float* __restrict__ out) {
    const int blk = blockIdx.x;          // b * kN + n
    const int b   = blk >> 9;
    const int t   = threadIdx.x;
    float* pooled = out + (size_t)blk * (kC * kPix);

    if (!keep[blk]) {
        for (int i = t; i < kC * kPix; i += 256) pooled[i] = 0.0f;
        return;
    }

    const int lvl = level[blk];
    const int W   = 1024 >> (lvl + 1);     // 256,128,64,32
    const int HW  = W * W;
    const float inv_stride = 1.0f / (float)(2 << lvl);
    const float* fp = (lvl == 1) ? p4 : (lvl == 2) ? p8 : (lvl == 3) ? p16 : p32;
    fp += (size_t)b * kC * HW;

    const float* vv = verts + (size_t)blk * 4;
    const float rx1 = vv[0] * inv_stride;
    const float ry1 = vv[1] * inv_stride;
    const float rx2 = vv[2] * inv_stride;
    const float ry2 = vv[3] * inv_stride;
    const float bw = (rx2 - rx1) * (1.0f / (float)kOut);
    const float bh = (ry2 - ry1) * (1.0f / (float)kOut);

    __shared__ __align__(16) float s_tile[kChk * kSlot];       // zero-padded tiles
    __shared__ __align__(16) float s_temp[kChk * 16 * kMaxTWp];// per-wave Wy*tile
    __shared__ __align__(16) float s_Wy[16 * kWyPitch];        // row-major, pitch 132
    __shared__ __align__(16) float s_WxT[kMaxTWp * 16];        // [k][n], pitch 16
    __shared__ int   s_xi0[kS], s_xi1[kS], s_yi0[kS], s_yi1[kS];
    __shared__ float s_lx[kS], s_ly[kS];

    if (t < kS) {
        float g  = ((float)t + 0.5f) * 0.5f;
        float sx = fminf(fmaxf(rx1 + g * bw, 0.0f), (float)(W - 1));
        float fx = floorf(sx);
        int   xi = (int)fx;
        s_xi0[t] = xi;
        s_xi1[t] = min(xi + 1, W - 1);
        s_lx[t]  = sx - fx;
        float sy = fminf(fmaxf(ry1 + g * bh, 0.0f), (float)(W - 1));
        float fy = floorf(sy);
        int   yi = (int)fy;
        s_yi0[t] = yi;
        s_yi1[t] = min(yi + 1, W - 1);
        s_ly[t]  = sy - fy;
    }
    __syncthreads();

    const int tx0 = s_xi0[0], tx1 = s_xi1[kS - 1];
    const int ty0 = s_yi0[0], ty1 = s_yi1[kS - 1];
    const int TW  = tx1 - tx0 + 1;
    const int TH  = ty1 - ty0 + 1;
    const int TWp = (TW + 15) & ~15;
    const int THp = (TH + 3) & ~3;
    const bool use_wmma = (TWp <= kMaxTWp) && (THp <= kMaxTHp) &&
                          (TWp * THp <= kSlot);

    if (use_wmma) {
        // zero tiles + weight matrices (pads must be exact zeros)
        for (int i = t; i < kChk * kSlot; i += 256) s_tile[i] = 0.0f;
        for (int i = t; i < 16 * kWyPitch; i += 256) s_Wy[i] = 0.0f;
        for (int i = t; i < kMaxTWp * 16; i += 256) s_WxT[i] = 0.0f;
        __syncthreads();

        // build separable weights (pooling 0.25 folded into Wy)
        if (t < kOut) {
            float* row = s_Wy + t * kWyPitch;
            for (int sub = 0; sub < 2; ++sub) {
                int iy = 2 * t + sub;
                float lyv = s_ly[iy];
                row[s_yi0[iy] - ty0] += 0.25f * (1.0f - lyv);
                row[s_yi1[iy] - ty0] += 0.25f * lyv;
            }
        } else if (t >= 32 && t < 32 + kOut) {
            int n = t - 32;
            for (int sub = 0; sub < 2; ++sub) {
                int ix = 2 * n + sub;
                float lxv = s_lx[ix];
                s_WxT[(s_xi0[ix] - tx0) * 16 + n] += 1.0f - lxv;
                s_WxT[(s_xi1[ix] - tx0) * 16 + n] += lxv;
            }
        }
        __syncthreads();

        const int tileN = TW * TH;
        const int wv = t >> 5;          // wave id == channel-in-chunk
        const int ln = t & 31;
        const int h2 = ln >> 4;         // half of wave
        const int am = ln & 15;         // A-row m / B-col n
        const float* tl = s_tile + wv * kSlot;
        float*       tp = s_temp + wv * (16 * kMaxTWp);

        for (int c0 = 0; c0 < kC; c0 += kChk) {
            // ---- async stage: 8 channel tiles -> LDS (zero-padded pitch) --
            for (int e = t; e < kChk * tileN; e += 256) {
                int ch = e / tileN;
                int o  = e - ch * tileN;
                int ty = o / TW;
                int tx = o - ty * TW;
                const float* ga = fp + (size_t)(c0 + ch) * HW
                                     + (size_t)(ty0 + ty) * W + (tx0 + tx);
                uint32_t la = (uint32_t)(uintptr_t)(s_tile + ch * kSlot + ty * TWp + tx);
                asm volatile("global_load_async_to_lds_b32 %0, %1, off"
                             :: "v"(la), "v"(ga) : "memory");
            }
            asm volatile("s_wait_asynccnt 0" ::: "memory");
            __syncthreads();

            // ---- stage 1: temp(16xTWp) = Wy * tile ----------------------
            for (int n0 = 0; n0 < TWp; n0 += 16) {
                v8f acc = {};
                for (int k0 = 0; k0 < THp; k0 += 4) {
                    int kk = k0 + 2 * h2;
                    v2f A = *(const v2f*)(s_Wy + am * kWyPitch + kk);
                    v2f Bf;
                    Bf.x = tl[kk * TWp + n0 + am];
                    Bf.y = tl[(kk + 1) * TWp + n0 + am];
                    acc = __builtin_amdgcn_wmma_f32_16x16x4_f32(
                        false, A, false, Bf, (short)0, acc, false, false);
                }
#pragma unroll
                for (int r = 0; r < 8; ++r)
                    tp[(r + 8 * h2) * TWp + n0 + am] = acc[r];
            }

            // ---- stage 2: out(16x16) = temp * Wx^T ----------------------
            v8f acc2 = {};
            for (int k0 = 0; k0 < TWp; k0 += 4) {
                int kk = k0 + 2 * h2;
                v2f A = *(const v2f*)(tp + am * TWp + kk);
                v2f Bf;
                Bf.x = s_WxT[kk * 16 + am];
                Bf.y = s_WxT[(kk + 1) * 16 + am];
                acc2 = __builtin_amdgcn_wmma_f32_16x16x4_f32(
                    false, A, false, Bf, (short)0, acc2, false, false);
            }
            float* po = pooled + (size_t)(c0 + wv) * kPix;
#pragma unroll
            for (int r = 0; r < 8; ++r) {
                int m2 = r + 8 * h2;
                if (m2 < kOut && am < kOut) po[m2 * kOut + am] = acc2[r];
            }
            __syncthreads();   // tile slots reused next round
        }
    } else {
        // -------- direct-global scalar fallback (elongated tiles) ---------
        for (int w = t; w < kC * kPix; w += 256) {
            int ch = w / kPix;
            int p  = w - ch * kPix;
            int oy = p / kOut;
            int ox = p - oy * kOut;
            const float* tlg = fp + (size_t)ch * HW;
            float acc = 0.0f;
#pragma unroll
            for (int sub = 0; sub < 4; ++sub) {
                int iy = oy * 2 + (sub >> 1);
                int ix = ox * 2 + (sub & 1);
                float lx = s_lx[ix], ly = s_ly[iy];
                int xa = s_xi0[ix], xb = s_xi1[ix];
                int ya = s_yi0[iy] * W, yb = s_yi1[iy] * W;
                acc += tlg[ya + xa] * (1.0f - ly) * (1.0f - lx)
                     + tlg[ya + xb] * (1.0f - ly) * lx
                     + tlg[yb + xa] * ly * (1.0f - lx)
                     + tlg[yb + xb] * ly * lx;
            }
            pooled[(size_t)ch * kPix + p] = acc * 0.25f;
        }
    }
}

// ---------------------------------------------------------------- launch
extern "C" void kernel_launch(void* const* d_in, const int* in_sizes, int n_in,
                              void* d_out, int out_size, void* d_ws, size_t ws_size,
                              hipStream_t stream) {
    const float* p4     = (const float*)d_in[0];
    const float* p8     = (const float*)d_in[1];
    const float* p16    = (const float*)d_in[2];
    const float* p32    = (const float*)d_in[3];
    const float* boxes  = (const float*)d_in[4];
    const float* scores = (const float*)d_in[5];

    float* out        = (float*)d_out;
    float* out_pooled = out;                                          // B*N*C*196
    float* out_boxes  = out + (size_t)kB * kN * kC * kPix;            // B*N*4
    float* out_keep   = out_boxes + (size_t)kB * kN * 4;              // B*N

    float* ws_verts = (float*)d_ws;                                   // B*N*4 f32
    int*   ws_level = (int*)((char*)d_ws + (size_t)kB * kN * 4 * sizeof(float));
    int*   ws_keep  = (int*)((char*)ws_level + (size_t)kB * kN * sizeof(int));

    roi_prep_kernel<<<dim3((kB * kN + 255) / 256), dim3(256), 0, stream>>>(
        boxes, ws_verts, ws_level);

    nms_kernel<<<dim3(kB), dim3(kN), 0, stream>>>(
        scores, ws_verts, ws_level, ws_keep, out_boxes, out_keep);

    roi_align_kernel<<<dim3(kB * kN), dim3(256), 0, stream>>>(
        p4, p8, p16, p32, ws_verts, ws_level, ws_keep, out_pooled);
}